// Mask_77283641524292
// MI455X (gfx1250) — compile-verified
//
#include <hip/hip_runtime.h>
#include <stdint.h>

// ---------------------------------------------------------------------------
// Top-K masking for N = 2^25, K = N/100 on MI455X (gfx1250).
// Bandwidth-bound (~0.03 flop/byte). 3 streaming passes ~= 0.9 GB traffic
// -> ~38us floor at 23.3 TB/s HBM; passes 2-3 partially served by 192MB L2.
// gfx1250 path: double-buffered GLOBAL_LOAD_ASYNC_TO_LDS_B128 staging in the
// histogram and write kernels, tracked with ASYNCcnt (s_wait_asynccnt).
// No WMMA: the workload has no matrix op; forcing one would waste cycles.
// ---------------------------------------------------------------------------

#define HIST_BINS   4096
#define HIST_BLOCKS 512
#define HIST_TPB    256
#define NBLK_COUNT  2048
#define CAND_MAX    131072

// workspace layout (u32 words)
#define OFF_HIST   0                       // 4096
#define OFF_CANDC  4096                    // 1
#define OFF_SEL    4100                    // [0]=B [1]=K' [2]=kcut [3]=icut
#define OFF_BLKC   4104                    // 2048 (written, not accumulated -> no zeroing)
#define OFF_BLKO   (4104 + NBLK_COUNT)     // 2048
#define OFF_CKEY   8448                    // CAND_MAX
#define OFF_CIDX   (8448 + CAND_MAX)       // CAND_MAX
#define ZERO_WORDS 4104                    // hist + candCount (+sel scratch)

// Monotone key: larger float -> larger unsigned key.
__device__ __forceinline__ unsigned monokey(float f) {
  unsigned b = __float_as_uint(f);
  return (b & 0x80000000u) ? ~b : (b | 0x80000000u);
}

// Low 32 bits of a generic LDS pointer == byte offset in the wave's LDS.
__device__ __forceinline__ unsigned lds_off(const void* p) {
  return (unsigned)(uintptr_t)p;
}

// gfx1250 async global->LDS copy (GVS mode: saddr base + vgpr byte offset).
__device__ __forceinline__ void issue_pair(unsigned ldsA, unsigned ldsG,
                                           unsigned byteOff,
                                           uint64_t actB, uint64_t gateB) {
  asm volatile("global_load_async_to_lds_b128 %0, %1, %2"
               :: "v"(ldsA), "v"(byteOff), "s"(actB) : "memory");
  asm volatile("global_load_async_to_lds_b128 %0, %1, %2"
               :: "v"(ldsG), "v"(byteOff), "s"(gateB) : "memory");
}
__device__ __forceinline__ void wait_async_le2() {
  asm volatile("s_wait_asynccnt 2" ::: "memory");
}
__device__ __forceinline__ void wait_async_0() {
  asm volatile("s_wait_asynccnt 0" ::: "memory");
}

// K1: zero scratch counters/histogram (graph-safe; no hipMemset).
__global__ void topk_zero_ws(unsigned* ws, int nwords) {
  int i = blockIdx.x * blockDim.x + threadIdx.x;
  if (i < nwords) ws[i] = 0u;
}

// K2: 4096-bin histogram of top-12 key bits, async-LDS double-buffered stage.
__global__ void __launch_bounds__(HIST_TPB)
topk_hist(const float* __restrict__ act, const float* __restrict__ gate,
          unsigned* __restrict__ hist, int n) {
  __shared__ unsigned h[HIST_BINS];
  __shared__ float4 stA[2][HIST_TPB];
  __shared__ float4 stG[2][HIST_TPB];

  const int t = threadIdx.x;
  for (int i = t; i < HIST_BINS; i += HIST_TPB) h[i] = 0u;
  __syncthreads();

  const uint64_t actB  = (uint64_t)act;
  const uint64_t gateB = (uint64_t)gate;
  const unsigned ldsA0 = lds_off(&stA[0][t]);
  const unsigned ldsA1 = lds_off(&stA[1][t]);
  const unsigned ldsG0 = lds_off(&stG[0][t]);
  const unsigned ldsG1 = lds_off(&stG[1][t]);

  const int nChunks = n / (HIST_TPB * 4);   // 1024 elements per block-chunk
  const int stride  = gridDim.x;

  int c = blockIdx.x;
  {
    unsigned off = ((unsigned)c * (HIST_TPB * 4u) + (unsigned)t * 4u) * 4u;
    issue_pair(ldsA0, ldsG0, off, actB, gateB);
  }
  int buf = 0;
  for (; c < nChunks; c += stride, buf ^= 1) {
    int nc = c + stride;
    if (nc < nChunks) {
      unsigned off = ((unsigned)nc * (HIST_TPB * 4u) + (unsigned)t * 4u) * 4u;
      issue_pair(buf ? ldsA0 : ldsA1, buf ? ldsG0 : ldsG1, off, actB, gateB);
      wait_async_le2();   // current buffer's 2 loads done (in-order completion)
    } else {
      wait_async_0();
    }
    float4 a4 = stA[buf][t];
    float4 g4 = stG[buf][t];
    atomicAdd(&h[monokey(a4.x * g4.x) >> 20], 1u);
    atomicAdd(&h[monokey(a4.y * g4.y) >> 20], 1u);
    atomicAdd(&h[monokey(a4.z * g4.z) >> 20], 1u);
    atomicAdd(&h[monokey(a4.w * g4.w) >> 20], 1u);
  }
  __syncthreads();
  for (int i = t; i < HIST_BINS; i += HIST_TPB)
    if (h[i]) atomicAdd(&hist[i], h[i]);
}

// K3: top-down scan -> threshold bin B and residual K'.
__global__ void topk_find_bin(const unsigned* __restrict__ hist,
                              unsigned* __restrict__ sel, int K) {
  if (threadIdx.x == 0 && blockIdx.x == 0) {
    unsigned cum = 0, B = 0, Kp = (unsigned)K;
    for (int b = HIST_BINS - 1; b >= 0; --b) {
      unsigned c = hist[b];
      if (cum + c >= (unsigned)K) { B = (unsigned)b; Kp = (unsigned)K - cum; break; }
      cum += c;
    }
    sel[0] = B; sel[1] = Kp;
  }
}

// K4: fused candidate extraction + per-block sure-winner (bin > B) counts.
// Contiguous per-block ranges so the counts feed deterministic compaction.
__global__ void __launch_bounds__(256)
topk_cand_count(const float* __restrict__ act, const float* __restrict__ gate,
                const unsigned* __restrict__ sel,
                unsigned* __restrict__ candCount,
                unsigned* __restrict__ ckey, unsigned* __restrict__ cidx,
                unsigned* __restrict__ blkCount, int perBlk) {
  __shared__ unsigned red[256];
  const unsigned B = sel[0];
  const int t = threadIdx.x;
  const int base = blockIdx.x * perBlk;
  unsigned sure = 0;
  for (int cb = 0; cb < perBlk; cb += 1024) {
    const int i0 = base + cb + t * 4;
    float4 a4 = *reinterpret_cast<const float4*>(act + i0);
    float4 g4 = *reinterpret_cast<const float4*>(gate + i0);
    const float av[4] = {a4.x * g4.x, a4.y * g4.y, a4.z * g4.z, a4.w * g4.w};
#pragma unroll
    for (int s = 0; s < 4; ++s) {
      unsigned key = monokey(av[s]);
      unsigned bin = key >> 20;
      if (bin > B) {
        sure++;
      } else if (bin == B) {
        unsigned pos = atomicAdd(candCount, 1u);
        if (pos < CAND_MAX) { ckey[pos] = key; cidx[pos] = (unsigned)(i0 + s); }
      }
    }
  }
  red[t] = sure; __syncthreads();
  for (int d = 128; d > 0; d >>= 1) {
    if (t < d) red[t] += red[t + d];
    __syncthreads();
  }
  if (t == 0) blkCount[blockIdx.x] = red[0];
}

// K5: refine to exact 32-bit cutoff key + deterministic index tie-break.
__global__ void __launch_bounds__(1024)
topk_refine(const unsigned* __restrict__ candCountP,
            const unsigned* __restrict__ ckey, const unsigned* __restrict__ cidx,
            unsigned* __restrict__ sel, int n) {
  __shared__ unsigned h2[1024];
  __shared__ unsigned sS1, sR1, sS2, sR2, sE, sLo, sHi, sCnt;
  const int t = threadIdx.x;
  unsigned C = *candCountP; if (C > CAND_MAX) C = CAND_MAX;
  const unsigned B = sel[0], r0 = sel[1];

  // Round A: key bits [19:10]
  h2[t] = 0u; __syncthreads();
  for (unsigned i = t; i < C; i += 1024u)
    atomicAdd(&h2[(ckey[i] >> 10) & 1023u], 1u);
  __syncthreads();
  if (t == 0) {
    unsigned cum = 0, s1 = 0, r1 = r0;
    for (int b = 1023; b >= 0; --b) {
      unsigned c = h2[b];
      if (cum + c >= r0) { s1 = (unsigned)b; r1 = r0 - cum; break; }
      cum += c;
    }
    sS1 = s1; sR1 = r1;
  }
  __syncthreads();
  const unsigned s1 = sS1, r1 = sR1;

  // Round B: key bits [9:0] among matching prefix
  h2[t] = 0u; __syncthreads();
  for (unsigned i = t; i < C; i += 1024u) {
    unsigned k = ckey[i];
    if (((k >> 10) & 1023u) == s1) atomicAdd(&h2[k & 1023u], 1u);
  }
  __syncthreads();
  if (t == 0) {
    unsigned cum = 0, s2 = 0, r2 = r1;
    for (int b = 1023; b >= 0; --b) {
      unsigned c = h2[b];
      if (cum + c >= r1) { s2 = (unsigned)b; r2 = r1 - cum; break; }
      cum += c;
    }
    sS2 = s2; sR2 = r2; sE = h2[s2];
  }
  __syncthreads();
  const unsigned kcut = (B << 20) | (s1 << 10) | sS2;
  const unsigned r2 = sR2, E = sE;

  unsigned icut;
  if (r2 >= E) {
    icut = 0xFFFFFFFFu;                 // all exact-key ties win
  } else {
    // smallest-index ties win: minimal m with count(key==kcut && idx<=m) >= r2
    if (t == 0) { sLo = 0u; sHi = (unsigned)(n - 1); }
    __syncthreads();
    for (;;) {
      unsigned lo = sLo, hi = sHi;
      if (lo >= hi) break;
      unsigned mid = lo + (hi - lo) / 2u;
      if (t == 0) sCnt = 0u;
      __syncthreads();
      unsigned local = 0;
      for (unsigned i = t; i < C; i += 1024u)
        if (ckey[i] == kcut && cidx[i] <= mid) local++;
      if (local) atomicAdd(&sCnt, local);
      __syncthreads();
      if (t == 0) { if (sCnt >= r2) sHi = mid; else sLo = mid + 1u; }
      __syncthreads();
    }
    icut = sLo;
  }
  if (t == 0) { sel[2] = kcut; sel[3] = icut; }
}

// K5b: add threshold-bin tie winners into per-block counts (tiny).
__global__ void __launch_bounds__(1024)
topk_tie_add(const unsigned* __restrict__ candCountP,
             const unsigned* __restrict__ ckey, const unsigned* __restrict__ cidx,
             const unsigned* __restrict__ sel, unsigned* __restrict__ blkCount,
             int perBlk) {
  unsigned C = *candCountP; if (C > CAND_MAX) C = CAND_MAX;
  const unsigned kcut = sel[2], icut = sel[3];
  for (unsigned i = threadIdx.x; i < C; i += 1024u) {
    unsigned k = ckey[i], id = cidx[i];
    if (k > kcut || (k == kcut && id <= icut))
      atomicAdd(&blkCount[id / (unsigned)perBlk], 1u);
  }
}

// K6: exclusive scan of per-block counts (tiny, serial is fine).
__global__ void topk_scan(const unsigned* __restrict__ blkCount,
                          unsigned* __restrict__ blkOff) {
  if (threadIdx.x == 0 && blockIdx.x == 0) {
    unsigned run = 0;
    for (int b = 0; b < NBLK_COUNT; ++b) { blkOff[b] = run; run += blkCount[b]; }
  }
}

// K7: masked output + index-ordered winner compaction, async-LDS staged.
// Next tile streams in under ASYNCcnt while the block runs its prefix scan.
__global__ void __launch_bounds__(256)
topk_write(const float* __restrict__ act, const float* __restrict__ gate,
           const unsigned* __restrict__ sel, const unsigned* __restrict__ blkOff,
           float* __restrict__ out, int* __restrict__ outIdx,
           int perBlk, int K) {
  __shared__ float4 stA[2][256];
  __shared__ float4 stG[2][256];
  __shared__ unsigned pfx[256];
  __shared__ unsigned sRun;
  const unsigned kcut = sel[2], icut = sel[3];
  const int t = threadIdx.x;
  const int base = blockIdx.x * perBlk;
  if (t == 0) sRun = blkOff[blockIdx.x];

  const uint64_t actB  = (uint64_t)act;
  const uint64_t gateB = (uint64_t)gate;
  const unsigned ldsA0 = lds_off(&stA[0][t]);
  const unsigned ldsA1 = lds_off(&stA[1][t]);
  const unsigned ldsG0 = lds_off(&stG[0][t]);
  const unsigned ldsG1 = lds_off(&stG[1][t]);

  // prime chunk 0
  {
    unsigned off = (unsigned)(base + t * 4) * 4u;
    issue_pair(ldsA0, ldsG0, off, actB, gateB);
  }
  __syncthreads();   // sRun visible

  int buf = 0;
  for (int cb = 0; cb < perBlk; cb += 1024, buf ^= 1) {
    const int nb = cb + 1024;
    if (nb < perBlk) {
      unsigned off = (unsigned)(base + nb + t * 4) * 4u;
      issue_pair(buf ? ldsA0 : ldsA1, buf ? ldsG0 : ldsG1, off, actB, gateB);
      wait_async_le2();
    } else {
      wait_async_0();
    }
    const int i0 = base + cb + t * 4;
    float4 a4 = stA[buf][t];
    float4 g4 = stG[buf][t];
    const float av[4] = {a4.x * g4.x, a4.y * g4.y, a4.z * g4.z, a4.w * g4.w};
    bool w[4]; unsigned cnt = 0;
#pragma unroll
    for (int s = 0; s < 4; ++s) {
      unsigned key = monokey(av[s]);
      w[s] = (key > kcut) || (key == kcut && (unsigned)(i0 + s) <= icut);
      cnt += w[s] ? 1u : 0u;
    }
    float4 o;
    o.x = w[0] ? av[0] : 0.0f;
    o.y = w[1] ? av[1] : 0.0f;
    o.z = w[2] ? av[2] : 0.0f;
    o.w = w[3] ? av[3] : 0.0f;
    *reinterpret_cast<float4*>(out + i0) = o;

    pfx[t] = cnt;
    __syncthreads();
    for (int d = 1; d < 256; d <<= 1) {      // Hillis-Steele inclusive scan
      unsigned v = (t >= d) ? pfx[t - d] : 0u;
      __syncthreads();
      pfx[t] += v;
      __syncthreads();
    }
    const unsigned incl = pfx[t];
    const unsigned total = pfx[255];
    const unsigned runBase = sRun;
    unsigned pos = runBase + incl - cnt;     // exclusive offset for this thread
#pragma unroll
    for (int s = 0; s < 4; ++s) {
      if (w[s]) {
        if (pos < (unsigned)K) outIdx[pos] = i0 + s;
        pos++;
      }
    }
    __syncthreads();
    if (t == 0) sRun = runBase + total;
    __syncthreads();
  }
}

extern "C" void kernel_launch(void* const* d_in, const int* in_sizes, int n_in,
                              void* d_out, int out_size, void* d_ws, size_t ws_size,
                              hipStream_t stream) {
  (void)n_in; (void)out_size; (void)ws_size;
  const float* act  = (const float*)d_in[0];
  const float* gate = (const float*)d_in[1];
  const int n = in_sizes[0];                // 33554432
  const int K = (int)((double)n * 0.01);    // 335544, matches int(N*0.01)

  unsigned* ws = (unsigned*)d_ws;
  float* out   = (float*)d_out;
  int* outIdx  = (int*)((float*)d_out + n); // win_ind region (flat concat)

  const int perBlk = n / NBLK_COUNT;        // 16384, exact for N = 2^25

  topk_zero_ws<<<(ZERO_WORDS + 255) / 256, 256, 0, stream>>>(ws, ZERO_WORDS);
  topk_hist<<<HIST_BLOCKS, HIST_TPB, 0, stream>>>(act, gate, ws + OFF_HIST, n);
  topk_find_bin<<<1, 1, 0, stream>>>(ws + OFF_HIST, ws + OFF_SEL, K);
  topk_cand_count<<<NBLK_COUNT, 256, 0, stream>>>(act, gate, ws + OFF_SEL,
                                                  ws + OFF_CANDC, ws + OFF_CKEY,
                                                  ws + OFF_CIDX, ws + OFF_BLKC,
                                                  perBlk);
  topk_refine<<<1, 1024, 0, stream>>>(ws + OFF_CANDC, ws + OFF_CKEY,
                                      ws + OFF_CIDX, ws + OFF_SEL, n);
  topk_tie_add<<<1, 1024, 0, stream>>>(ws + OFF_CANDC, ws + OFF_CKEY,
                                       ws + OFF_CIDX, ws + OFF_SEL,
                                       ws + OFF_BLKC, perBlk);
  topk_scan<<<1, 1, 0, stream>>>(ws + OFF_BLKC, ws + OFF_BLKO);
  topk_write<<<NBLK_COUNT, 256, 0, stream>>>(act, gate, ws + OFF_SEL,
                                             ws + OFF_BLKO, out, outIdx,
                                             perBlk, K);
}